// ComplexExtraction_52862457480044
// MI455X (gfx1250) — compile-verified
//
#include <hip/hip_runtime.h>
#include <math.h>

// ---------------------------------------------------------------------------
// Complex f32 GEMM on CDNA5 WMMA (V_WMMA_F32_16X16X4_F32), wave32.
// D = A(complex) x B(complex) [*scale +mask +bias, crelu], batched-strided.
// A fragment (16x4 f32): lane<16 holds row M=lane, {K=k0,k0+1}; lane>=16 holds
// row M=lane-16, {K=k0+2,k0+3}.  B fragment mirrors A with M->N.
// f32 WMMA has no A/B NEG (ISA 7.12: NEG = CNeg only), so complex product uses
// three accumulators: P=Ar*Br, M=Ai*Bi, I=Ar*Bi+Ai*Br; re=P-M, im=I.
// BKN=0 ([N,K] B): NT=2 col-subtiles/wave, 3x global_load_b128 + 8 wmma/step.
// BKN=1 ([K,N] B): NT=1 (AV GEMM has N=64 -> all 4 waves useful),
//                  1x b128 + 2x b64 + 4 wmma/step, 3 pointer bumps.
// ---------------------------------------------------------------------------

typedef __attribute__((ext_vector_type(2))) float v2f;
typedef __attribute__((ext_vector_type(8))) float v8f;

struct GemmP {
  const float2* A; const float2* Bm; float2* C;
  const float2* mask; const float2* bias;
  int M, N, K;
  int lda, ldb, ldc;                       // row strides in complex elements
  long sA_b, sA_h, sB_b, sB_h, sC_b, sC_h, sM_b, sM_h;  // z = b*H + h strides
  int H;                                   // heads per batch for z decomposition
  int b_kn;                                // 0: B is [N,K]; 1: B is [K,N]
  float scale;
  int crelu;
  int fscatter;                            // final [B,T,E*C] -> [B,E,C,T] scatter
  int To, Co, Eo;
};

__device__ __forceinline__ void cmac4(v8f& pr, v8f& mi, v8f& im,
                                      v2f ar, v2f ai, v2f br, v2f bi) {
  pr = __builtin_amdgcn_wmma_f32_16x16x4_f32(false, ar, false, br, (short)0, pr, false, false);
  mi = __builtin_amdgcn_wmma_f32_16x16x4_f32(false, ai, false, bi, (short)0, mi, false, false);
  im = __builtin_amdgcn_wmma_f32_16x16x4_f32(false, ar, false, bi, (short)0, im, false, false);
  im = __builtin_amdgcn_wmma_f32_16x16x4_f32(false, ai, false, br, (short)0, im, false, false);
}

template <int BKN>
__global__ __launch_bounds__(128) void cgemm_wmma_f32(GemmP p) {
  constexpr int NT = (BKN == 0) ? 2 : 1;     // 16-col subtiles per wave
  const int lane = threadIdx.x & 31;
  const int wave = threadIdx.x >> 5;
  const int half = lane >> 4;       // 0 or 1
  const int l16  = lane & 15;
  const int z    = blockIdx.z;
  const int zb   = z / p.H;
  const int zh   = z - zb * p.H;

  const float2* A = p.A  + (size_t)zb * p.sA_b + (size_t)zh * p.sA_h;
  const float2* B = p.Bm + (size_t)zb * p.sB_b + (size_t)zh * p.sB_h;

  const int m0 = blockIdx.y * 16;
  const int n0 = blockIdx.x * (64 * NT) + wave * (16 * NT);

  // Clamp row/col indices: OOB A-rows / B-cols only affect D rows/cols that
  // are masked at store time, so clamping (no zero-fill) is sufficient.
  const int mA  = m0 + l16;
  const int mAc = (mA < p.M) ? mA : (p.M - 1);
  const int nB0 = n0 + l16;
  const int nc0 = (nB0 < p.N) ? nB0 : (p.N - 1);
  const int nB1 = nB0 + 16;
  const int nc1 = (nB1 < p.N) ? nB1 : (p.N - 1);

  v8f pr0 = {}, mi0 = {}, im0 = {};
  v8f pr1 = {}, mi1 = {}, im1 = {};

  const int K4 = p.K & ~3;

  // ---- pointer-bumped main loop (all lda/ldb are even; kk starts at 2*half)
  {
    const float4* __restrict__ ap =
        reinterpret_cast<const float4*>(A) + (((size_t)mAc * p.lda) >> 1) + half;

    if (BKN == 0) {
      const float4* __restrict__ bp0 =
          reinterpret_cast<const float4*>(B) + (((size_t)nc0 * p.ldb) >> 1) + half;
      const float4* __restrict__ bp1 =
          reinterpret_cast<const float4*>(B) + (((size_t)nc1 * p.ldb) >> 1) + half;
#pragma unroll 4
      for (int k0 = 0; k0 < K4; k0 += 4) {
        float4 av = *ap;  ap  += 2;
        float4 b0 = *bp0; bp0 += 2;
        float4 b1 = *bp1; bp1 += 2;
        v2f ar = {av.x, av.z}, ai = {av.y, av.w};
        v2f br0v = {b0.x, b0.z}, bi0v = {b0.y, b0.w};
        v2f br1v = {b1.x, b1.z}, bi1v = {b1.y, b1.w};
        cmac4(pr0, mi0, im0, ar, ai, br0v, bi0v);
        cmac4(pr1, mi1, im1, ar, ai, br1v, bi1v);
      }
    } else {
      const long step = 4L * p.ldb;
      const float2* __restrict__ q00 = B + (size_t)(2 * half) * p.ldb + nc0;
      const float2* __restrict__ q01 = B + (size_t)(2 * half + 1) * p.ldb + nc0;
#pragma unroll 4
      for (int k0 = 0; k0 < K4; k0 += 4) {
        float4 av = *ap; ap += 2;
        float2 b00 = *q00; q00 += step;
        float2 b01 = *q01; q01 += step;
        v2f ar = {av.x, av.z}, ai = {av.y, av.w};
        v2f br0v = {b00.x, b01.x}, bi0v = {b00.y, b01.y};
        cmac4(pr0, mi0, im0, ar, ai, br0v, bi0v);
      }
    }
  }

  if (K4 < p.K) {                      // K remainder: must zero-fill OOB k
    const int kk = K4 + 2 * half;
    v2f ar, ai;  ar.x = 0.f; ar.y = 0.f; ai.x = 0.f; ai.y = 0.f;
    if (kk < p.K)     { float2 t = A[(size_t)mAc * p.lda + kk];     ar.x = t.x; ai.x = t.y; }
    if (kk + 1 < p.K) { float2 t = A[(size_t)mAc * p.lda + kk + 1]; ar.y = t.x; ai.y = t.y; }
    v2f br0v, bi0v, br1v, bi1v;
    br0v.x = br0v.y = bi0v.x = bi0v.y = 0.f;
    br1v.x = br1v.y = bi1v.x = bi1v.y = 0.f;
    if (BKN == 0) {
      if (kk < p.K) {
        float2 t = B[(size_t)nc0 * p.ldb + kk]; br0v.x = t.x; bi0v.x = t.y;
        t = B[(size_t)nc1 * p.ldb + kk];        br1v.x = t.x; bi1v.x = t.y;
      }
      if (kk + 1 < p.K) {
        float2 t = B[(size_t)nc0 * p.ldb + kk + 1]; br0v.y = t.x; bi0v.y = t.y;
        t = B[(size_t)nc1 * p.ldb + kk + 1];        br1v.y = t.x; bi1v.y = t.y;
      }
    } else {
      if (kk < p.K) {
        float2 t = B[(size_t)kk * p.ldb + nc0]; br0v.x = t.x; bi0v.x = t.y;
      }
      if (kk + 1 < p.K) {
        float2 t = B[((size_t)kk + 1) * p.ldb + nc0]; br0v.y = t.x; bi0v.y = t.y;
      }
    }
    cmac4(pr0, mi0, im0, ar, ai, br0v, bi0v);
    if (NT == 2) cmac4(pr1, mi1, im1, ar, ai, br1v, bi1v);
  }

  // Epilogue: scale, +mask, +bias, crelu, store (or final scatter).
  const float2* Mz = nullptr;
  if (p.mask) Mz = p.mask + (size_t)zb * p.sM_b + (size_t)zh * p.sM_h;
  float2* Cz = p.C + (p.fscatter ? 0 : ((size_t)zb * p.sC_b + (size_t)zh * p.sC_h));

#pragma unroll
  for (int t = 0; t < NT; ++t) {
    const int n = n0 + t * 16 + l16;
    if (n >= p.N) continue;
    v8f aP = t ? pr1 : pr0;
    v8f aM = t ? mi1 : mi0;
    v8f aI = t ? im1 : im0;
    float bre = 0.f, bim = 0.f;
    if (p.bias) { float2 bb = p.bias[n]; bre = bb.x; bim = bb.y; }
    for (int r = 0; r < 8; ++r) {
      const int m = m0 + half * 8 + r;
      if (m >= p.M) break;
      float re  = (aP[r] - aM[r]) * p.scale + bre;
      float imv = aI[r] * p.scale + bim;
      if (Mz) { float2 mk = Mz[(size_t)m * p.ldc + n]; re += mk.x; imv += mk.y; }
      if (p.crelu) { re = fmaxf(re, 0.f); imv = fmaxf(imv, 0.f); }
      if (!p.fscatter) {
        Cz[(size_t)m * p.ldc + n] = make_float2(re, imv);
      } else {
        const int b = m / p.To, tt = m - b * p.To;
        const int e = n / p.Co, c = n - e * p.Co;
        Cz[(((size_t)(b * p.Eo + e)) * p.Co + c) * (size_t)p.To + tt] = make_float2(re, imv);
      }
    }
  }
}

// ---------------------------------------------------------------------------
// Elementwise kernels
// ---------------------------------------------------------------------------

// x[b,t,c*E+e] = emb[b,e,c,t]
__global__ void k_xflat(const float2* __restrict__ emb, float2* __restrict__ x,
                        int E, int C, int T, long total) {
  long i = (long)blockIdx.x * blockDim.x + threadIdx.x;
  if (i >= total) return;
  int e = (int)(i % E); long r = i / E;
  int c = (int)(r % C); r /= C;
  int t = (int)(r % T); int b = (int)(r / T);
  x[i] = emb[(((size_t)b * E + e) * C + c) * T + t];
}

// kv[b,t,c] = src[b,c]
__global__ void k_tile_kv(const float2* __restrict__ src, float2* __restrict__ dst,
                          int T, int C, long total) {
  long i = (long)blockIdx.x * blockDim.x + threadIdx.x;
  if (i >= total) return;
  int c = (int)(i % C);
  long bt = i / C;
  int b = (int)(bt / T);
  dst[i] = src[(size_t)b * C + c];
}

// Per-batch sum(|x|), sum(|x|^2): one block per batch, grid-stride in block.
__global__ void k_gln_reduce(const float2* __restrict__ x, float* __restrict__ stats,
                             long nper) {
  const int b = blockIdx.x;
  const float2* xb = x + (size_t)b * nper;
  float s = 0.f, s2 = 0.f;
  for (long i = threadIdx.x; i < nper; i += blockDim.x) {
    float2 vv = xb[i];
    float m = sqrtf(vv.x * vv.x + vv.y * vv.y);
    s += m; s2 = fmaf(m, m, s2);
  }
  __shared__ float sh[512];
  sh[threadIdx.x] = s; sh[256 + threadIdx.x] = s2;
  __syncthreads();
  for (int st = 128; st > 0; st >>= 1) {
    if (threadIdx.x < st) {
      sh[threadIdx.x] += sh[threadIdx.x + st];
      sh[256 + threadIdx.x] += sh[256 + threadIdx.x + st];
    }
    __syncthreads();
  }
  if (threadIdx.x == 0) { stats[2 * b] = sh[0]; stats[2 * b + 1] = sh[256]; }
}

// Normalize |x| globally per batch, keep phase, y = gamma[d1]*nx + beta[d1]
// (real beta adds to real part only, matching JAX real+complex semantics).
__global__ void k_gln_apply(float2* __restrict__ x, const float* __restrict__ stats,
                            const float* __restrict__ g, const float* __restrict__ bt,
                            long nper, int D2, long total) {
  long i = (long)blockIdx.x * blockDim.x + threadIdx.x;
  if (i >= total) return;
  int b = (int)(i / nper);
  long rem = i - (long)b * nper;
  int d1 = (int)(rem / D2);
  float mean = stats[2 * b] / (float)nper;
  float var  = stats[2 * b + 1] / (float)nper - mean * mean;
  float inv  = rsqrtf(var + 1e-8f);
  float2 vv = x[i];
  float mag = sqrtf(vv.x * vv.x + vv.y * vv.y);
  float nm = (mag - mean) * inv;
  float cs = 1.f, sn = 0.f;
  if (mag > 0.f) { float r = 1.f / mag; cs = vv.x * r; sn = vv.y * r; }
  x[i] = make_float2(g[d1] * nm * cs + bt[d1], g[d1] * nm * sn);
}

// ---------------------------------------------------------------------------
// Host orchestration
// ---------------------------------------------------------------------------

static inline GemmP mkgemm(const float2* A, const float2* B, float2* C,
                           int M, int N, int K, int lda, int ldb, int ldc) {
  GemmP p{};
  p.A = A; p.Bm = B; p.C = C; p.M = M; p.N = N; p.K = K;
  p.lda = lda; p.ldb = ldb; p.ldc = ldc;
  p.H = 1; p.scale = 1.f;
  return p;
}

static inline void rungemm(hipStream_t s, const GemmP& p, int Z) {
  if (p.b_kn) {
    dim3 g((p.N + 63) / 64, (p.M + 15) / 16, Z);      // NT=1: 4 waves x 16 cols
    hipLaunchKernelGGL((cgemm_wmma_f32<1>), g, dim3(128), 0, s, p);
  } else {
    dim3 g((p.N + 127) / 128, (p.M + 15) / 16, Z);    // NT=2: 4 waves x 32 cols
    hipLaunchKernelGGL((cgemm_wmma_f32<0>), g, dim3(128), 0, s, p);
  }
}

extern "C" void kernel_launch(void* const* d_in, const int* in_sizes, int n_in,
                              void* d_out, int out_size, void* d_ws, size_t ws_size,
                              hipStream_t stream) {
  (void)in_sizes; (void)n_in; (void)out_size;
  constexpr int Bc = 2, Ec = 16, Cc = 512, Tc = 626, Hh = 8, HD = 64;
  const float SCALE = 0.125f;               // 64^-0.5
  const long  MR = (long)Bc * Tc;           // 1252 rows for [B*T, ...] GEMMs

  // d_in order = setup_inputs() recursive dict insertion order.
  const float2* emb  = (const float2*)d_in[0];
  const float2* hrtf = (const float2*)d_in[1];
  auto P2 = [&](int i) { return (const float2*)d_in[i]; };
  auto PF = [&](int i) { return (const float*)d_in[i]; };
  // 2:fc1_w 3:fc1_b 4..11:mix(wq,bq,wk,bk,wv,bv,wo,bo) 12..19:hrtf_attn
  // 20:fc_hrtf_w 21:fc_hrtf_b 22:g_hrtf 23:b_hrtf 24..31:cross
  // 32:g_bn 33:b_bn 34:out_w 35:out_b  then bn{i}@36+10i, g{i}@44+10i, b{i}@45+10i

  // --- workspace bump allocator ---
  char* base = (char*)d_ws;
  size_t off = 0;
  auto alloc = [&](size_t bytes) -> char* {
    char* p = base + off;
    off += (bytes + 255) & ~(size_t)255;
    return p;
  };
  const size_t xflatB = (size_t)MR * 8192 * sizeof(float2);         // 82.05 MB
  const size_t awB    = (size_t)Bc * Hh * Tc * Tc * sizeof(float2); // 50.15 MB
  float2* xflat = (float2*)alloc(xflatB);
  float2* aw0   = xflat;                 // alias: xflat dead after fc1
  float2* aw1   = (float2*)alloc(awB);
  float2* x  = (float2*)alloc((size_t)MR * Cc * sizeof(float2));
  float2* q  = (float2*)alloc((size_t)MR * Cc * sizeof(float2));
  float2* k  = (float2*)alloc((size_t)MR * Cc * sizeof(float2));
  float2* v  = (float2*)alloc((size_t)MR * Cc * sizeof(float2));
  float2* kv = (float2*)alloc((size_t)MR * Cc * sizeof(float2));
  float2* t0 = (float2*)alloc((size_t)MR * Cc * sizeof(float2));
  float2* t1 = (float2*)alloc((size_t)MR * Cc * sizeof(float2));
  float2* hq   = (float2*)alloc((size_t)4 * Cc * sizeof(float2));
  float2* hk   = (float2*)alloc((size_t)4 * Cc * sizeof(float2));
  float2* hv   = (float2*)alloc((size_t)4 * Cc * sizeof(float2));
  float2* ho   = (float2*)alloc((size_t)4 * Cc * sizeof(float2));
  float2* hbuf = (float2*)alloc((size_t)4 * Cc * sizeof(float2));
  float2* hsc  = (float2*)alloc((size_t)Bc * Hh * 2 * 2 * sizeof(float2));
  float2* hfc  = (float2*)alloc((size_t)Bc * Cc * sizeof(float2));
  float*  stats = (float*)alloc(64);
  if (off > ws_size) return;             // insufficient scratch: bail deterministically

  auto proj = [&](const float2* src, int M, int wi, float2* dst) {
    GemmP p = mkgemm(src, P2(wi), dst, M, Cc, Cc, Cc, Cc, Cc);
    p.bias = P2(wi + 1);
    rungemm(stream, p, 1);
  };

  // Full-size attention on [B,T,C] tensors; kv source may differ from q source.
  auto attn = [&](const float2* qsrc, const float2* kvsrc, int pbase,
                  float2* awdst, const float2* mask, float2* outdst) {
    proj(qsrc, (int)MR, pbase + 0, q);
    proj(kvsrc, (int)MR, pbase + 2, k);
    proj(kvsrc, (int)MR, pbase + 4, v);
    // scores: per (b,h)  [T,HD] x [T,HD]^T -> [T,T], *SCALE (+mask), crelu
    GemmP s = mkgemm(q, k, awdst, Tc, Tc, HD, Cc, Cc, Tc);
    s.sA_b = (long)Tc * Cc; s.sA_h = HD;
    s.sB_b = (long)Tc * Cc; s.sB_h = HD;
    s.sC_b = (long)Hh * Tc * Tc; s.sC_h = (long)Tc * Tc;
    s.H = Hh; s.scale = SCALE; s.crelu = 1;
    if (mask) { s.mask = mask; s.sM_b = s.sC_b; s.sM_h = s.sC_h; }
    rungemm(stream, s, Bc * Hh);
    // AV: [T,T] x [T,HD] -> [T,HD]   (B in [K,N] layout)
    GemmP a = mkgemm(awdst, v, t0, Tc, HD, Tc, Tc, Cc, Cc);
    a.sA_b = (long)Hh * Tc * Tc; a.sA_h = (long)Tc * Tc;
    a.sB_b = (long)Tc * Cc; a.sB_h = HD;
    a.sC_b = (long)Tc * Cc; a.sC_h = HD;
    a.H = Hh; a.b_kn = 1;
    rungemm(stream, a, Bc * Hh);
    proj(t0, (int)MR, pbase + 6, outdst);
  };

  const long NE = 256;

  // 1) input transpose: emb [B,E,C,T] -> xflat [B*T, C*E]
  {
    long tot = MR * 8192;
    hipLaunchKernelGGL(k_xflat, dim3((unsigned)((tot + NE - 1) / NE)), dim3(NE), 0, stream,
                       emb, xflat, Ec, Cc, Tc, tot);
  }
  // 2) fc1: [1252,8192] x [8192,512]^T + b -> x
  {
    GemmP p = mkgemm(xflat, P2(2), x, (int)MR, Cc, Ec * Cc, Ec * Cc, Ec * Cc, Cc);
    p.bias = P2(3);
    rungemm(stream, p, 1);
  }
  // 3) mix self-attention (x -> x)
  attn(x, x, 4, aw0, nullptr, x);

  // 4) hrtf self-attention ([B,2,512], tiny GEMMs)
  {
    proj(hrtf, 4, 12, hq);
    proj(hrtf, 4, 14, hk);
    proj(hrtf, 4, 16, hv);
    GemmP s = mkgemm(hq, hk, hsc, 2, 2, HD, Cc, Cc, 2);
    s.sA_b = 2L * Cc; s.sA_h = HD;
    s.sB_b = 2L * Cc; s.sB_h = HD;
    s.sC_b = (long)Hh * 4; s.sC_h = 4;
    s.H = Hh; s.scale = SCALE; s.crelu = 1;
    rungemm(stream, s, Bc * Hh);
    GemmP a = mkgemm(hsc, hv, ho, 2, HD, 2, 2, Cc, Cc);
    a.sA_b = (long)Hh * 4; a.sA_h = 4;
    a.sB_b = 2L * Cc; a.sB_h = HD;
    a.sC_b = 2L * Cc; a.sC_h = HD;
    a.H = Hh; a.b_kn = 1;
    rungemm(stream, a, Bc * Hh);
    proj(ho, 4, 18, hbuf);
  }
  // 5) globln(hbuf) over dims (2,512) per batch; gamma/beta indexed by dim of size 2
  {
    long nper = 2L * Cc, tot = (long)Bc * nper;
    hipLaunchKernelGGL(k_gln_reduce, dim3(Bc), dim3(256), 0, stream, hbuf, stats, nper);
    hipLaunchKernelGGL(k_gln_apply, dim3((unsigned)((tot + NE - 1) / NE)), dim3(NE), 0, stream,
                       hbuf, stats, PF(22), PF(23), nper, Cc, tot);
  }
  // 6) fc_hrtf: [B,1024] x [1024,512]^T + b -> hfc [B,512]
  {
    GemmP p = mkgemm(hbuf, P2(20), hfc, Bc, Cc, 2 * Cc, 2 * Cc, 2 * Cc, Cc);
    p.bias = P2(21);
    rungemm(stream, p, 1);
  }
  // 7) kv = tile(hfc) to [B,T,512]
  {
    long tot = MR * Cc;
    hipLaunchKernelGGL(k_tile_kv, dim3((unsigned)((tot + NE - 1) / NE)), dim3(NE), 0, stream,
                       hfc, kv, Tc, Cc, tot);
  }
  // 8) cross-attention (Q from x, KV from kv) -> t1 ; its aw is discarded
  attn(x, kv, 24, aw0, nullptr, t1);
  // 9) globln(t1) with g_bn/b_bn (gamma/beta indexed by T)
  {
    long nper = (long)Tc * Cc, tot = (long)Bc * nper;
    hipLaunchKernelGGL(k_gln_reduce, dim3(Bc), dim3(256), 0, stream, t1, stats, nper);
    hipLaunchKernelGGL(k_gln_apply, dim3((unsigned)((tot + NE - 1) / NE)), dim3(NE), 0, stream,
                       t1, stats, PF(32), PF(33), nper, Cc, tot);
  }
  // 10) four bn self-attentions with attn-weight mask chaining + globln
  {
    const float2* mask = nullptr;
    for (int i = 0; i < 4; ++i) {
      float2* dst = (i & 1) ? aw0 : aw1;     // never equal to `mask`
      attn(t1, t1, 36 + 10 * i, dst, mask, t1);
      mask = dst;
      long nper = (long)Tc * Cc, tot = (long)Bc * nper;
      hipLaunchKernelGGL(k_gln_reduce, dim3(Bc), dim3(256), 0, stream, t1, stats, nper);
      hipLaunchKernelGGL(k_gln_apply, dim3((unsigned)((tot + NE - 1) / NE)), dim3(NE), 0, stream,
                         t1, stats, PF(44 + 10 * i), PF(45 + 10 * i), nper, Cc, tot);
    }
  }
  // 11) out: [1252,512] x [512,8192]^T + b, scattered directly to [B,E,C,T]
  {
    GemmP p = mkgemm(t1, P2(34), (float2*)d_out, (int)MR, Ec * Cc, Cc, Cc, Cc, 0);
    p.bias = P2(35);
    p.fscatter = 1; p.To = Tc; p.Co = Cc; p.Eo = Ec;
    rungemm(stream, p, 1);
  }
}